// ContextAwarePathIntegrator_84413287235615
// MI455X (gfx1250) — compile-verified
//
#include <hip/hip_runtime.h>
#include <hip/hip_bf16.h>
#include <math.h>

typedef float v2f __attribute__((ext_vector_type(2)));
typedef float v8f __attribute__((ext_vector_type(8)));

#define PI_F 3.14159265358979323846f

__device__ __forceinline__ float relu1(float x) {
  // single-instruction ReLU; fmaxf/fmed3 both lower to 2 VALU ops (canonicalize+max)
  float r;
  asm("v_max_num_f32 %0, 0, %1" : "=v"(r) : "v"(x));
  return r;
}

// One wave (32 lanes) owns a 32-row batch tile and integrates one reset-
// delimited time segment. Every lane owns one batch row (carry ct in regs);
// the per-step MLP runs as two 16-row WMMA tiles so EXEC stays full everywhere.
__global__ __launch_bounds__(32) void cap_scan_kernel(
    const float* __restrict__ t_in,
    const float* __restrict__ action,
    const int*   __restrict__ reset_mask,
    const float* __restrict__ W_a,
    const float* __restrict__ b_a,
    const float* __restrict__ W1,
    const float* __restrict__ b1,
    const float* __restrict__ W2,
    const float* __restrict__ b2,
    float* __restrict__ out,
    int B, int T)
{
  const int t0 = blockIdx.y;
  // Only segment-start blocks do work: t==0 or reset step (uniform branch).
  if (t0 != 0 && reset_mask[t0] != 0) return;

  const int  lane = threadIdx.x;      // 0..31, owns batch row base+lane
  const int  n    = lane & 15;        // N-column within a 16-wide tile
  const bool hi   = lane >= 16;
  const int  koff = hi ? 2 : 0;       // f32 A-fragment: lanes 16..31 hold K+2,K+3
  const int  base = blockIdx.x * 32;
  int row = base + lane;
  const bool rowvalid = row < B;
  if (!rowvalid) row = B - 1;

  __shared__ float pose_lds[32 * 12];   // 32 rows x K=12 (9 padded to 12)
  __shared__ float h_lds[32 * 32];      // hidden activations, row-major
  __shared__ float p_lds[32 * 4];       // MLP outputs (3 used)

  // ---- constant B-matrix fragments, hoisted out of the scan ----
  // B 4x16 f32 fragment layout assumed: v.x = row K(=4kk+koff), v.y = row K+1,
  // lanes 0..15 carry column n with K,K+1; lanes 16..31 K+2,K+3 (mirrors A).
  v2f w1f[2][3];
  #pragma unroll
  for (int nt = 0; nt < 2; ++nt)
    #pragma unroll
    for (int kk = 0; kk < 3; ++kk) {
      int k0  = 4 * kk + koff;
      int col = nt * 16 + n;
      w1f[nt][kk].x = (k0     < 9) ? W1[(k0    ) * 32 + col] : 0.f;
      w1f[nt][kk].y = (k0 + 1 < 9) ? W1[(k0 + 1) * 32 + col] : 0.f;
    }
  v2f w2f[8];
  #pragma unroll
  for (int kk = 0; kk < 8; ++kk) {
    int k0 = 4 * kk + koff;
    w2f[kk].x = (n < 3) ? W2[(k0    ) * 3 + n] : 0.f;
    w2f[kk].y = (n < 3) ? W2[(k0 + 1) * 3 + n] : 0.f;
  }

  // persistent bias broadcasts, fed as the C operand of the first WMMA of each
  // accumulation chain (D = A*B + bias) so no per-step register copies occur
  v8f bias_lo, bias_hi, bias_p;
  {
    const float b1l = b1[n], b1h = b1[16 + n];
    const float b2n = (n < 3) ? b2[n] : 0.f;
    #pragma unroll
    for (int j = 0; j < 8; ++j) { bias_lo[j] = b1l; bias_hi[j] = b1h; bias_p[j] = b2n; }
  }

  float wa[24];
  #pragma unroll
  for (int i = 0; i < 24; ++i) wa[i] = W_a[i];
  const float ba0 = b_a[0], ba1 = b_a[1], ba2 = b_a[2];

  // zero the K=9..11 pose padding once (each lane pads its own row)
  pose_lds[lane * 12 + 9]  = 0.f;
  pose_lds[lane * 12 + 10] = 0.f;
  pose_lds[lane * 12 + 11] = 0.f;

  // carry: each lane holds its row's 3x3 (row-major)
  float ct[9];
  #pragma unroll
  for (int i = 0; i < 9; ++i) ct[i] = 0.f;
  if (t0 == 0) {
    const float* p0 = t_in + (size_t)row * T * 9;   // ct0 = t_in[:,0]
    #pragma unroll
    for (int i = 0; i < 9; ++i) ct[i] = p0[i];
  }
  __syncthreads();

  for (int t = t0; t < T; ++t) {
    if (t > t0 && reset_mask[t] == 0) break;        // next segment's block
    const bool reset = (reset_mask[t] == 0);        // block-uniform

    // delta = action[row,t] @ W_a + b_a (every lane, its own row)
    const float* ap = action + ((size_t)row * T + t) * 8;
    float d0 = ba0, d1 = ba1, d2 = ba2;
    #pragma unroll
    for (int a = 0; a < 8; ++a) {
      float av = ap[a];
      d0 = fmaf(av, wa[a * 3 + 0], d0);
      d1 = fmaf(av, wa[a * 3 + 1], d1);
      d2 = fmaf(av, wa[a * 3 + 2], d2);
    }
    const float dr = d0 * PI_F, dxi = d1, dyi = d2;
    if (t + 1 < T) __builtin_prefetch(ap + 8, 0, 0);   // global_prefetch_b8

    float rot, tx, ty;     // t_d parameters
    float E[9];            // ct_eff
    #pragma unroll
    for (int i = 0; i < 9; ++i) E[i] = ct[i];

    if (!reset) {
      // =============== context MLP via V_WMMA_F32_16X16X4_F32 =============
      __syncthreads();                  // WAR vs previous iteration LDS reads
      {
        float* pr = pose_lds + lane * 12;   // pose = [ct[0..5], dr, dx, dy]
        pr[0] = ct[0]; pr[1] = ct[1]; pr[2] = ct[2];
        pr[3] = ct[3]; pr[4] = ct[4]; pr[5] = ct[5];
        pr[6] = dr;    pr[7] = dxi;   pr[8] = dyi;
      }
      __syncthreads();

      // h = relu(pose @ W1 + b1): two 16-row M-tiles, 3 k-steps x 2 N-tiles
      v2f a0[3], a1[3];
      #pragma unroll
      for (int kk = 0; kk < 3; ++kk) {
        a0[kk].x = pose_lds[(n     ) * 12 + 4 * kk + koff];
        a0[kk].y = pose_lds[(n     ) * 12 + 4 * kk + koff + 1];
        a1[kk].x = pose_lds[(16 + n) * 12 + 4 * kk + koff];
        a1[kk].y = pose_lds[(16 + n) * 12 + 4 * kk + koff + 1];
      }
      v8f acc00 = __builtin_amdgcn_wmma_f32_16x16x4_f32(false, a0[0], false, w1f[0][0],
                                                        (short)0, bias_lo, false, false);
      v8f acc01 = __builtin_amdgcn_wmma_f32_16x16x4_f32(false, a0[0], false, w1f[1][0],
                                                        (short)0, bias_hi, false, false);
      v8f acc10 = __builtin_amdgcn_wmma_f32_16x16x4_f32(false, a1[0], false, w1f[0][0],
                                                        (short)0, bias_lo, false, false);
      v8f acc11 = __builtin_amdgcn_wmma_f32_16x16x4_f32(false, a1[0], false, w1f[1][0],
                                                        (short)0, bias_hi, false, false);
      #pragma unroll
      for (int kk = 1; kk < 3; ++kk) {
        acc00 = __builtin_amdgcn_wmma_f32_16x16x4_f32(false, a0[kk], false, w1f[0][kk],
                                                      (short)0, acc00, false, false);
        acc01 = __builtin_amdgcn_wmma_f32_16x16x4_f32(false, a0[kk], false, w1f[1][kk],
                                                      (short)0, acc01, false, false);
        acc10 = __builtin_amdgcn_wmma_f32_16x16x4_f32(false, a1[kk], false, w1f[0][kk],
                                                      (short)0, acc10, false, false);
        acc11 = __builtin_amdgcn_wmma_f32_16x16x4_f32(false, a1[kk], false, w1f[1][kk],
                                                      (short)0, acc11, false, false);
      }
      // relu + stage h row-major [m][32] for the next A-fragment gather
      #pragma unroll
      for (int j = 0; j < 8; ++j) {
        int mm = j + (hi ? 8 : 0);          // C/D layout: vgpr j -> M=j (+8 hi)
        h_lds[(mm     ) * 32 + n]      = relu1(acc00[j]);
        h_lds[(mm     ) * 32 + 16 + n] = relu1(acc01[j]);
        h_lds[(mm + 16) * 32 + n]      = relu1(acc10[j]);
        h_lds[(mm + 16) * 32 + 16 + n] = relu1(acc11[j]);
      }
      __syncthreads();

      // p = h @ W2 + b2: 16x32 @ 32x16(pad, 3 used), 8 k-steps per tile
      v2f c0, c1;
      c0.x = h_lds[(n     ) * 32 + koff];
      c0.y = h_lds[(n     ) * 32 + koff + 1];
      c1.x = h_lds[(16 + n) * 32 + koff];
      c1.y = h_lds[(16 + n) * 32 + koff + 1];
      v8f accp0 = __builtin_amdgcn_wmma_f32_16x16x4_f32(false, c0, false, w2f[0],
                                                        (short)0, bias_p, false, false);
      v8f accp1 = __builtin_amdgcn_wmma_f32_16x16x4_f32(false, c1, false, w2f[0],
                                                        (short)0, bias_p, false, false);
      #pragma unroll
      for (int kk = 1; kk < 8; ++kk) {
        v2f d0, d1;
        d0.x = h_lds[(n     ) * 32 + 4 * kk + koff];
        d0.y = h_lds[(n     ) * 32 + 4 * kk + koff + 1];
        d1.x = h_lds[(16 + n) * 32 + 4 * kk + koff];
        d1.y = h_lds[(16 + n) * 32 + 4 * kk + koff + 1];
        accp0 = __builtin_amdgcn_wmma_f32_16x16x4_f32(false, d0, false, w2f[kk],
                                                      (short)0, accp0, false, false);
        accp1 = __builtin_amdgcn_wmma_f32_16x16x4_f32(false, d1, false, w2f[kk],
                                                      (short)0, accp1, false, false);
      }
      if (n < 3) {
        #pragma unroll
        for (int j = 0; j < 8; ++j) {
          int mm = j + (hi ? 8 : 0);
          p_lds[(mm     ) * 4 + n] = accp0[j];
          p_lds[(mm + 16) * 4 + n] = accp1[j];
        }
      }
      __syncthreads();

      rot = p_lds[lane * 4 + 0];
      tx  = p_lds[lane * 4 + 1];
      ty  = p_lds[lane * 4 + 2];
      // E already = ct (ct_eff == ct on non-reset steps)
    } else {
      // reset step: t_d from action delta, ct_eff = t_in[:,t]
      rot = dr; tx = dxi; ty = dyi;
      const float* pi = t_in + ((size_t)row * T + t) * 9;
      #pragma unroll
      for (int i = 0; i < 9; ++i) E[i] = pi[i];
    }

    // t_out = ct_eff @ [[c,-s,tx],[s,c,ty],[0,0,1]]  (every lane, full EXEC)
    {
      float s, c;
      __sincosf(rot, &s, &c);
      float O[9];
      #pragma unroll
      for (int i = 0; i < 3; ++i) {
        float e0 = E[i * 3 + 0], e1 = E[i * 3 + 1], e2 = E[i * 3 + 2];
        O[i * 3 + 0] = fmaf(e0, c, e1 * s);
        O[i * 3 + 1] = fmaf(e1, c, -(e0 * s));
        O[i * 3 + 2] = fmaf(e0, tx, fmaf(e1, ty, e2));
      }
      #pragma unroll
      for (int i = 0; i < 9; ++i) ct[i] = O[i];
      if (rowvalid) {
        float* op = out + ((size_t)row * T + t) * 9;
        #pragma unroll
        for (int i = 0; i < 9; ++i) op[i] = O[i];
      }
    }
  }
}

extern "C" void kernel_launch(void* const* d_in, const int* in_sizes, int n_in,
                              void* d_out, int out_size, void* d_ws, size_t ws_size,
                              hipStream_t stream) {
  const float* t_in       = (const float*)d_in[0];
  const float* action     = (const float*)d_in[1];
  const int*   reset_mask = (const int*)  d_in[2];
  const float* W_a        = (const float*)d_in[3];
  const float* b_a        = (const float*)d_in[4];
  const float* W1         = (const float*)d_in[5];
  const float* b1         = (const float*)d_in[6];
  const float* W2         = (const float*)d_in[7];
  const float* b2         = (const float*)d_in[8];
  float*       out        = (float*)d_out;

  const int T = in_sizes[2];                 // reset_mask is (T,)
  const int B = in_sizes[0] / (9 * T);       // t_in is (B,T,3,3)

  dim3 grid((B + 31) / 32, T);
  dim3 block(32);
  cap_scan_kernel<<<grid, block, 0, stream>>>(t_in, action, reset_mask,
                                              W_a, b_a, W1, b1, W2, b2,
                                              out, B, T);
}